// JointWassersteinLoss_49151605735883
// MI455X (gfx1250) — compile-verified
//
#include <hip/hip_runtime.h>

#define Nn 4096
#define Mm 4096
#define Dd 512
#define NCc 10
#define EPSc 0.05f
#define NITER 20

#define BM 128
#define BN 64
#define KT 32

typedef __attribute__((ext_vector_type(2))) float v2f;
typedef __attribute__((ext_vector_type(8))) float v8f;
typedef __attribute__((ext_vector_type(4))) int v4i;

#ifndef __has_builtin
#define __has_builtin(x) 0
#endif
#if __has_builtin(__builtin_amdgcn_global_load_async_to_lds_b128)
#define HAVE_ASYNC_LDS 1
#else
#define HAVE_ASYNC_LDS 0
#endif

// global (AS1) / LDS (AS3) pointer casts for the async builtin
#define GQ(p) ((__attribute__((address_space(1))) v4i*)(p))
#define LQ(p) ((__attribute__((address_space(3))) v4i*)(p))

static __device__ __forceinline__ void wait_async0() {
#if HAVE_ASYNC_LDS
#if __has_builtin(__builtin_amdgcn_s_wait_asynccnt)
  __builtin_amdgcn_s_wait_asynccnt(0);
#else
  asm volatile("s_wait_asynccnt 0x0" ::: "memory");
#endif
#endif
}

static __device__ __forceinline__ float wave_sum(float x) {
  #pragma unroll
  for (int o = 16; o; o >>= 1) x += __shfl_down(x, o, 32);
  return x;
}
static __device__ __forceinline__ float wave_max(float x) {
  #pragma unroll
  for (int o = 16; o; o >>= 1) x = fmaxf(x, __shfl_down(x, o, 32));
  return x;
}

// stage one 128x32 A panel + 64x32 B panel into LDS (async if available)
static __device__ __forceinline__ void stage_tiles(
    float (*dstA)[33], float (*dstB)[33], const float* __restrict__ XP,
    const float* __restrict__ XQ, int m0, int n0, int k0, int t) {
#if HAVE_ASYNC_LDS
  #pragma unroll
  for (int i = 0; i < 4; ++i) {
    int idx = t + i * 256, row = idx >> 3, c4 = (idx & 7) << 2;
    __builtin_amdgcn_global_load_async_to_lds_b128(
        GQ(XP + (size_t)(m0 + row) * Dd + k0 + c4),
        LQ(&dstA[row][c4]), 0, 0);
  }
  #pragma unroll
  for (int i = 0; i < 2; ++i) {
    int idx = t + i * 256, row = idx >> 3, c4 = (idx & 7) << 2;
    __builtin_amdgcn_global_load_async_to_lds_b128(
        GQ(XQ + (size_t)(n0 + row) * Dd + k0 + c4),
        LQ(&dstB[row][c4]), 0, 0);
  }
#else
  #pragma unroll
  for (int i = 0; i < 4; ++i) {
    int idx = t + i * 256, row = idx >> 3, c4 = (idx & 7) << 2;
    float4 av = *(const float4*)(XP + (size_t)(m0 + row) * Dd + k0 + c4);
    dstA[row][c4 + 0] = av.x; dstA[row][c4 + 1] = av.y;
    dstA[row][c4 + 2] = av.z; dstA[row][c4 + 3] = av.w;
  }
  #pragma unroll
  for (int i = 0; i < 2; ++i) {
    int idx = t + i * 256, row = idx >> 3, c4 = (idx & 7) << 2;
    float4 bv = *(const float4*)(XQ + (size_t)(n0 + row) * Dd + k0 + c4);
    dstB[row][c4 + 0] = bv.x; dstB[row][c4 + 1] = bv.y;
    dstB[row][c4 + 2] = bv.z; dstB[row][c4 + 3] = bv.w;
  }
#endif
}

// ---- row norms of XP (D=512) and XQ: one wave per row --------------------
__global__ __launch_bounds__(256) void xnorms_kernel(
    const float* __restrict__ XP, const float* __restrict__ XQ,
    float* __restrict__ xxP, float* __restrict__ yyQ) {
  int gw = (blockIdx.x * 256 + threadIdx.x) >> 5;
  int lane = threadIdx.x & 31;
  const float* src;
  float* dst;
  int row;
  if (gw < Nn) { src = XP; dst = xxP; row = gw; }
  else         { src = XQ; dst = yyQ; row = gw - Nn; }
  const float* p = src + (size_t)row * Dd;
  float s = 0.f;
  for (int k = lane; k < Dd; k += 32) { float x = p[k]; s = fmaf(x, x, s); }
  s = wave_sum(s);
  if (lane == 0) dst[row] = s;
}

// ---- row norms of YP / YQ (NC=10): one thread per row --------------------
__global__ __launch_bounds__(256) void ynorms_kernel(
    const float* __restrict__ YP, const float* __restrict__ YQ,
    float* __restrict__ ynP, float* __restrict__ ynQ) {
  int i = blockIdx.x * 256 + threadIdx.x;
  const float* p;
  float* d;
  int row;
  if (i < Nn) { p = YP + (size_t)i * NCc;        d = ynP; row = i; }
  else        { p = YQ + (size_t)(i - Nn) * NCc; d = ynQ; row = i - Nn; }
  float s = 0.f;
  #pragma unroll
  for (int k = 0; k < NCc; ++k) s = fmaf(p[k], p[k], s);
  d[row] = s;
}

// ---- CX GEMM via V_WMMA_F32_16X16X4_F32, 128x64 block tile ---------------
// 256 threads (8 waves), each wave computes a 2x2 grid of 16x16 tiles.
// Double-buffered LDS staging via GLOBAL_LOAD_ASYNC_TO_LDS (if available).
__global__ __launch_bounds__(256) void gemm_cx_kernel(
    const float* __restrict__ XP, const float* __restrict__ XQ,
    const float* __restrict__ YP, const float* __restrict__ YQ,
    const float* __restrict__ xxP, const float* __restrict__ yyQ,
    const float* __restrict__ ynP, const float* __restrict__ ynQ,
    float* __restrict__ CX, unsigned* __restrict__ scal) {
  __shared__ float As[2][BM][33];
  __shared__ float Bs[2][BN][33];
  __shared__ float YPs[BM][NCc];
  __shared__ float YQs[BN][NCc];

  const int t = threadIdx.x;
  const int lane = t & 31, wv = t >> 5;
  const int m0 = blockIdx.y * BM, n0 = blockIdx.x * BN;

  if (t < BM) {
    #pragma unroll
    for (int k = 0; k < NCc; ++k) YPs[t][k] = YP[(size_t)(m0 + t) * NCc + k];
  }
  if (t < BN) {
    #pragma unroll
    for (int k = 0; k < NCc; ++k) YQs[t][k] = YQ[(size_t)(n0 + t) * NCc + k];
  }

  const int mBase = (wv & 3) * 32;   // 4 wave-rows cover 128 M
  const int nBase = (wv >> 2) * 32;  // 2 wave-cols cover 64 N
  const int ll = lane & 15, lh = lane >> 4;

  v8f acc[2][2] = {};

  stage_tiles(As[0], Bs[0], XP, XQ, m0, n0, 0, t);
  wait_async0();
  __syncthreads();

  #pragma unroll 1
  for (int k0 = 0; k0 < Dd; k0 += KT) {
    const int cur = (k0 >> 5) & 1;
    if (k0 + KT < Dd)
      stage_tiles(As[cur ^ 1], Bs[cur ^ 1], XP, XQ, m0, n0, k0 + KT, t);

    #pragma unroll
    for (int kk = 0; kk < KT; kk += 4) {
      // A 16x4 layout: lanes 0-15 hold K={0,1}, lanes 16-31 hold K={2,3}
      int ka = kk + 2 * lh;
      v2f a0, a1, b0, b1;
      a0.x = As[cur][mBase + ll][ka];       a0.y = As[cur][mBase + ll][ka + 1];
      a1.x = As[cur][mBase + 16 + ll][ka];  a1.y = As[cur][mBase + 16 + ll][ka + 1];
      b0.x = Bs[cur][nBase + ll][ka];       b0.y = Bs[cur][nBase + ll][ka + 1];
      b1.x = Bs[cur][nBase + 16 + ll][ka];  b1.y = Bs[cur][nBase + 16 + ll][ka + 1];
      acc[0][0] = __builtin_amdgcn_wmma_f32_16x16x4_f32(
          false, a0, false, b0, (short)0, acc[0][0], false, false);
      acc[0][1] = __builtin_amdgcn_wmma_f32_16x16x4_f32(
          false, a0, false, b1, (short)0, acc[0][1], false, false);
      acc[1][0] = __builtin_amdgcn_wmma_f32_16x16x4_f32(
          false, a1, false, b0, (short)0, acc[1][0], false, false);
      acc[1][1] = __builtin_amdgcn_wmma_f32_16x16x4_f32(
          false, a1, false, b1, (short)0, acc[1][1], false, false);
    }
    wait_async0();
    __syncthreads();
  }

  // epilogue: CX = max(xx+yy-2*dot,0); track max(CX), max(CY)
  float cxm = 0.f, cym = 0.f;
  #pragma unroll
  for (int mi = 0; mi < 2; ++mi) {
    #pragma unroll
    for (int nj = 0; nj < 2; ++nj) {
      int nloc = nBase + nj * 16 + ll;
      int n = n0 + nloc;
      float yq = yyQ[n], yqn = ynQ[n];
      #pragma unroll
      for (int r = 0; r < 8; ++r) {
        int mloc = mBase + mi * 16 + r + 8 * lh;  // lanes 16-31 -> M=r+8
        int m = m0 + mloc;
        float cx = fmaxf(xxP[m] + yq - 2.0f * acc[mi][nj][r], 0.0f);
        CX[(size_t)m * Mm + n] = cx;
        cxm = fmaxf(cxm, cx);
        float s = 0.f;
        #pragma unroll
        for (int k = 0; k < NCc; ++k) s = fmaf(YPs[mloc][k], YQs[nloc][k], s);
        float cy = fmaxf(ynP[m] + yqn - 2.0f * s, 0.0f);
        cym = fmaxf(cym, cy);
      }
    }
  }
  cxm = wave_max(cxm);
  cym = wave_max(cym);
  if (lane == 0) {
    atomicMax(&scal[0], __float_as_uint(cxm));  // values >= 0: uint order ok
    atomicMax(&scal[1], __float_as_uint(cym));
  }
}

// ---- C = CX + beta*CY (CY recomputed, K=10), track max(C) ----------------
__global__ __launch_bounds__(256) void build_c_kernel(
    float* __restrict__ C, const float* __restrict__ YP,
    const float* __restrict__ YQ, const float* __restrict__ ynP,
    const float* __restrict__ ynQ, unsigned* __restrict__ scal) {
  __shared__ float yp[NCc];
  int m = blockIdx.y;
  int n = blockIdx.x * 256 + threadIdx.x;
  if (threadIdx.x < NCc) yp[threadIdx.x] = YP[(size_t)m * NCc + threadIdx.x];
  __syncthreads();
  float cxm = __uint_as_float(scal[0]);
  float cym = __uint_as_float(scal[1]);
  float beta = (cym > 0.f) ? (cxm / fmaxf(cym, 1e-30f)) : 0.f;
  float s = 0.f;
  #pragma unroll
  for (int k = 0; k < NCc; ++k) s = fmaf(yp[k], YQ[(size_t)n * NCc + k], s);
  float cy = fmaxf(ynP[m] + ynQ[n] - 2.0f * s, 0.0f);
  size_t e = (size_t)m * Mm + n;
  float c = C[e] + beta * cy;
  C[e] = c;
  float cm = wave_max(c);
  if ((threadIdx.x & 31) == 0) atomicMax(&scal[2], __float_as_uint(cm));
}

// ---- K = exp(-C/reg), reg = EPS*max(C) -----------------------------------
__global__ __launch_bounds__(256) void build_k_kernel(
    const float* __restrict__ C, float* __restrict__ K,
    const unsigned* __restrict__ scal) {
  float cmax = __uint_as_float(scal[2]);
  float ninvreg = -1.0f / (EPSc * cmax);
  size_t i = ((size_t)blockIdx.x * 256 + threadIdx.x) * 4;
  float4 c4 = *(const float4*)(C + i);
  float4 k4;
  k4.x = __expf(c4.x * ninvreg);
  k4.y = __expf(c4.y * ninvreg);
  k4.z = __expf(c4.z * ninvreg);
  k4.w = __expf(c4.w * ninvreg);
  *(float4*)(K + i) = k4;
}

__global__ __launch_bounds__(256) void init_uv_kernel(float* u, float* v) {
  int i = blockIdx.x * 256 + threadIdx.x;
  u[i] = 1.0f / Nn;
  v[i] = 1.0f / Mm;
}

// ---- vsum += K^T u over a 128-row chunk (thread per column) --------------
__global__ __launch_bounds__(256) void ktu_kernel(
    const float* __restrict__ K, const float* __restrict__ u,
    float* __restrict__ vsum) {
  __shared__ float us[128];
  int t = threadIdx.x;
  int r0 = blockIdx.y * 128;
  if (t < 128) us[t] = u[r0 + t];
  __syncthreads();
  int c = blockIdx.x * 256 + t;
  const float* kp = K + (size_t)r0 * Mm + c;
  float acc = 0.f;
  #pragma unroll 8
  for (int r = 0; r < 128; ++r) acc = fmaf(kp[(size_t)r * Mm], us[r], acc);
  atomicAdd(&vsum[c], acc);
}

// ---- usum += K v over a 1024-col chunk (wave per row) --------------------
__global__ __launch_bounds__(256) void kv_kernel(
    const float* __restrict__ K, const float* __restrict__ v,
    float* __restrict__ usum) {
  __shared__ float vs[1024];
  int t = threadIdx.x;
  int c0 = blockIdx.x * 1024;
  #pragma unroll
  for (int i = 0; i < 4; ++i) vs[t + i * 256] = v[c0 + t + i * 256];
  __syncthreads();
  int lane = t & 31, wv = t >> 5;
  int r = blockIdx.y * 8 + wv;
  const float* kp = K + (size_t)r * Mm + c0;
  float acc = 0.f;
  #pragma unroll 8
  for (int k = 0; k < 32; ++k) {
    int cc = lane + 32 * k;
    acc = fmaf(kp[cc], vs[cc], acc);
  }
  acc = wave_sum(acc);
  if (lane == 0) atomicAdd(&usum[r], acc);
}

// ---- dst = marg / sum; also zero the other accumulator -------------------
__global__ __launch_bounds__(256) void div_zero_kernel(
    float* __restrict__ dst, const float* __restrict__ sum, float marg,
    float* __restrict__ zbuf) {
  int i = blockIdx.x * 256 + threadIdx.x;
  dst[i] = marg / sum[i];
  zbuf[i] = 0.f;
}

// ---- out += sum(C * u K v) ----------------------------------------------
__global__ __launch_bounds__(256) void loss_kernel(
    const float* __restrict__ C, const float* __restrict__ K,
    const float* __restrict__ u, const float* __restrict__ v,
    float* __restrict__ out) {
  int m = blockIdx.y;
  int base = blockIdx.x * 1024 + threadIdx.x * 4;
  float um = u[m];
  size_t e = (size_t)m * Mm + base;
  float4 c4 = *(const float4*)(C + e);
  float4 k4 = *(const float4*)(K + e);
  float4 v4 = *(const float4*)(v + base);
  float acc = um * (c4.x * k4.x * v4.x + c4.y * k4.y * v4.y +
                    c4.z * k4.z * v4.z + c4.w * k4.w * v4.w);
  acc = wave_sum(acc);
  if ((threadIdx.x & 31) == 0) atomicAdd(out, acc);
}

extern "C" void kernel_launch(void* const* d_in, const int* in_sizes, int n_in,
                              void* d_out, int out_size, void* d_ws,
                              size_t ws_size, hipStream_t stream) {
  // setup_inputs order: XQ [M,D], YQ [M,NC], XP [N,D], YP [N,NC]
  const float* XQ = (const float*)d_in[0];
  const float* YQ = (const float*)d_in[1];
  const float* XP = (const float*)d_in[2];
  const float* YP = (const float*)d_in[3];
  float* out = (float*)d_out;

  const size_t NM = (size_t)Nn * Mm;
  float* ws = (float*)d_ws;
  float* Cbuf = ws;            // N*M  (CX, then C in-place)
  float* Kbuf = ws + NM;       // N*M
  float* xxP = Kbuf + NM;      // N
  float* yyQ = xxP + Nn;       // M
  float* ynP = yyQ + Mm;       // N
  float* ynQ = ynP + Nn;       // M
  float* u = ynQ + Mm;         // N
  float* v = u + Nn;           // M
  float* usum = v + Mm;        // N
  float* vsum = usum + Nn;     // M
  unsigned* scal = (unsigned*)(vsum + Mm);  // [0]=cx_max [1]=cy_max [2]=c_max

  (void)hipMemsetAsync(scal, 0, 3 * sizeof(unsigned), stream);
  (void)hipMemsetAsync(usum, 0, (size_t)(Nn + Mm) * sizeof(float), stream);
  (void)hipMemsetAsync(out, 0, sizeof(float), stream);

  xnorms_kernel<<<(Nn + Mm) / 8, 256, 0, stream>>>(XP, XQ, xxP, yyQ);
  ynorms_kernel<<<(Nn + Mm) / 256, 256, 0, stream>>>(YP, YQ, ynP, ynQ);

  gemm_cx_kernel<<<dim3(Mm / BN, Nn / BM), 256, 0, stream>>>(
      XP, XQ, YP, YQ, xxP, yyQ, ynP, ynQ, Cbuf, scal);

  build_c_kernel<<<dim3(Mm / 256, Nn), 256, 0, stream>>>(Cbuf, YP, YQ, ynP,
                                                         ynQ, scal);
  build_k_kernel<<<(unsigned)(NM / 1024), 256, 0, stream>>>(Cbuf, Kbuf, scal);
  init_uv_kernel<<<Nn / 256, 256, 0, stream>>>(u, v);

  for (int it = 0; it < NITER; ++it) {
    ktu_kernel<<<dim3(Mm / 256, Nn / 128), 256, 0, stream>>>(Kbuf, u, vsum);
    div_zero_kernel<<<Mm / 256, 256, 0, stream>>>(v, vsum, 1.0f / Mm, usum);
    kv_kernel<<<dim3(Mm / 1024, Nn / 8), 256, 0, stream>>>(Kbuf, v, usum);
    div_zero_kernel<<<Nn / 256, 256, 0, stream>>>(u, usum, 1.0f / Nn, vsum);
  }

  loss_kernel<<<dim3(Mm / 1024, Nn), 256, 0, stream>>>(Cbuf, Kbuf, u, v, out);
}